// Model_17265768530008
// MI455X (gfx1250) — compile-verified
//
#include <hip/hip_runtime.h>

#define HIDDEN 64
#define EIGD   16
#define NUSERS 100000
#define NITEMS 200000
#define NNODES 300000
#define NEDGES 1250000
#define NPATHS 14
#define LN_EPS 1e-5f

typedef __attribute__((ext_vector_type(2))) float v2f;
typedef __attribute__((ext_vector_type(8))) float v8f;

// ---- monotone float <-> uint encoding so unsigned atomicMax == float max ----
static __device__ __forceinline__ unsigned fenc(float f) {
  unsigned u = __float_as_uint(f);
  return (u & 0x80000000u) ? ~u : (u | 0x80000000u);
}
static __device__ __forceinline__ float fdec(unsigned u) {
  return (u & 0x80000000u) ? __uint_as_float(u ^ 0x80000000u)
                           : __uint_as_float(~u);
}

// ---- LayerNorm via matrix-pipe reduction ----------------------------------
// One wave32 per 16 nodes. A (16x4 f32, documented layout): lane L<16 holds
// node L's elements K0/K1, lane L+16 holds K2/K3 of node L-16. B = all-ones
// (layout-invariant), so after 16 accumulating V_WMMA_F32_16X16X4_F32 steps
// D[m][n] = sum_k x[m][k] for every n. Same trick with squared A gives sum x^2.
// Stats read back from documented D positions via readlane broadcasts.
__global__ void ln_wmma_kernel(const float* __restrict__ srcA,
                               const float* __restrict__ srcB,
                               int nA, float* __restrict__ y) {
  int lane = threadIdx.x & 31;
  int wave = threadIdx.x >> 5;
  int base = (blockIdx.x * (blockDim.x >> 5) + wave) << 4;  // first of 16 nodes
  if (base >= NNODES) return;  // wave-uniform: NNODES % 16 == 0

  int m = lane & 15;           // node within group handled by this lane
  int h = lane >> 4;           // K-half per documented A layout
  int node = base + m;
  const float* srow = (node < nA) ? (srcA + (size_t)node * HIDDEN)
                                  : (srcB + (size_t)(node - nA) * HIDDEN);
  const float* sp = srow + 2 * h;          // elements 4c+2h, 4c+2h+1
  float*       dp = y + (size_t)node * HIDDEN + 2 * h;

  const v2f ones = {1.0f, 1.0f};
  v8f accS = {};
  v8f accQ = {};
  float2 vbuf[16];
#pragma unroll
  for (int c = 0; c < 16; ++c) {
    float2 v = *(const float2*)(sp + 4 * c);
    vbuf[c] = v;
    v2f a  = {v.x, v.y};
    v2f a2 = {v.x * v.x, v.y * v.y};
    accS = __builtin_amdgcn_wmma_f32_16x16x4_f32(false, a,  false, ones,
                                                 (short)0, accS, false, false);
    accQ = __builtin_amdgcn_wmma_f32_16x16x4_f32(false, a2, false, ones,
                                                 (short)0, accQ, false, false);
  }

  // D[m][0] lives at: m<8 -> lane 0, acc[m];  m>=8 -> lane 16, acc[m-8].
  float s = 0.0f, q = 0.0f;
#pragma unroll
  for (int i = 0; i < 8; ++i) {
    float sLo = __shfl(accS[i], 0, 32);
    float sHi = __shfl(accS[i], 16, 32);
    float qLo = __shfl(accQ[i], 0, 32);
    float qHi = __shfl(accQ[i], 16, 32);
    s = (m == i)     ? sLo : s;
    s = (m == 8 + i) ? sHi : s;
    q = (m == i)     ? qLo : q;
    q = (m == 8 + i) ? qHi : q;
  }

  float mu  = s * (1.0f / HIDDEN);
  float var = q * (1.0f / HIDDEN) - mu * mu;
  float rin = rsqrtf(var + LN_EPS);
#pragma unroll
  for (int c = 0; c < 16; ++c) {
    float2 o = make_float2((vbuf[c].x - mu) * rin, (vbuf[c].y - mu) * rin);
    *(float2*)(dp + 4 * c) = o;
  }
}

// ---- per-edge scores: 16 lanes per edge (float4 each), segment max via atomicMax ----
__global__ void score_kernel(const float* __restrict__ y,
                             const float* __restrict__ eigs,
                             const int* __restrict__ row,
                             const int* __restrict__ col,
                             const int* __restrict__ pt,
                             const float* __restrict__ lambda0_l,
                             const float* __restrict__ pemb_l,
                             float* __restrict__ s0,
                             unsigned* __restrict__ m0u,
                             unsigned* __restrict__ m1u) {
  int e = blockIdx.x * (blockDim.x >> 4) + (threadIdx.x >> 4);
  if (e >= NEDGES) return;
  int lane = threadIdx.x & 15;
  int r = row[e], c = col[e];
  // warm the eig lines (gfx1250 global_prefetch_b8) while y gathers issue
  __builtin_prefetch(eigs + (size_t)r * EIGD, 0, 1);
  __builtin_prefetch(eigs + (size_t)c * EIGD, 0, 1);
  float4 a4 = ((const float4*)(y + (size_t)r * HIDDEN))[lane];
  float4 b4 = ((const float4*)(y + (size_t)c * HIDDEN))[lane];
  float p = 0.125f * (a4.x * b4.x + a4.y * b4.y + a4.z * b4.z + a4.w * b4.w);
  float er = eigs[(size_t)r * EIGD + lane];
  float ec = eigs[(size_t)c * EIGD + lane];
  float explam = expf(lambda0_l[0]);
  p += explam * er * ec;
#pragma unroll
  for (int off = 8; off > 0; off >>= 1) p += __shfl_xor(p, off, 16);
  if (lane == 0) {
    s0[e] = p;
    atomicMax(m0u + r, fenc(p));
    atomicMax(m1u + r, fenc(pemb_l[pt[e]]));
  }
}

// ---- exp(s - m), segment-sum denominators via L2 f32 atomics ----
__global__ void expsum_kernel(const float* __restrict__ s0,
                              const int* __restrict__ row,
                              const int* __restrict__ pt,
                              const float* __restrict__ pemb_l,
                              const unsigned* __restrict__ m0u,
                              const unsigned* __restrict__ m1u,
                              float* __restrict__ e0,
                              float* __restrict__ d0,
                              float* __restrict__ d1) {
  int e = blockIdx.x * blockDim.x + threadIdx.x;
  if (e >= NEDGES) return;
  int r = row[e];
  float e0v = expf(s0[e] - fdec(m0u[r]));
  e0[e] = e0v;
  atomicAdd(d0 + r, e0v);
  float e1v = expf(pemb_l[pt[e]] - fdec(m1u[r]));
  atomicAdd(d1 + r, e1v);
}

// ---- final attention coefficient a = 0.5*(e0/d0 + e1/d1); overwrites s0 buffer ----
__global__ void coef_kernel(const int* __restrict__ row,
                            const int* __restrict__ pt,
                            const float* __restrict__ pemb_l,
                            const unsigned* __restrict__ m1u,
                            const float* __restrict__ e0,
                            const float* __restrict__ d0,
                            const float* __restrict__ d1,
                            float* __restrict__ a) {
  int e = blockIdx.x * blockDim.x + threadIdx.x;
  if (e >= NEDGES) return;
  int r = row[e];
  float a0 = e0[e] / d0[r];
  float a1 = expf(pemb_l[pt[e]] - fdec(m1u[r])) / d1[r];
  a[e] = 0.5f * (a0 + a1);
}

// ---- scatter: 64 threads per edge, coalesced y[col] read, L2-resident atomic adds ----
__global__ void aggregate_kernel(const float* __restrict__ a,
                                 const float* __restrict__ y,
                                 const int* __restrict__ row,
                                 const int* __restrict__ col,
                                 float* __restrict__ out) {
  int idx = blockIdx.x * blockDim.x + threadIdx.x;  // exactly NEDGES*64
  int e = idx >> 6;
  int c = idx & (HIDDEN - 1);
  float av = a[e];
  int rr = row[e];
  int cc = col[e];
  atomicAdd(out + (size_t)rr * HIDDEN + c, av * y[(size_t)cc * HIDDEN + c]);
}

// ---- elementwise helpers ----
__global__ void init_out_kernel(const float* __restrict__ ue,
                                const float* __restrict__ ie,
                                float* __restrict__ out) {
  int i = blockIdx.x * blockDim.x + threadIdx.x;  // exactly NNODES*HIDDEN
  out[i] = (i < NUSERS * HIDDEN) ? ue[i] : ie[i - NUSERS * HIDDEN];
}
__global__ void add_kernel(float* __restrict__ out, const float* __restrict__ x) {
  int i = blockIdx.x * blockDim.x + threadIdx.x;
  out[i] += x[i];
}
__global__ void scale_kernel(float* __restrict__ out, float s) {
  int i = blockIdx.x * blockDim.x + threadIdx.x;
  out[i] *= s;
}

extern "C" void kernel_launch(void* const* d_in, const int* in_sizes, int n_in,
                              void* d_out, int out_size, void* d_ws, size_t ws_size,
                              hipStream_t stream) {
  (void)in_sizes; (void)n_in; (void)out_size; (void)ws_size;
  const float* ue      = (const float*)d_in[0];
  const float* ie      = (const float*)d_in[1];
  const float* eigs    = (const float*)d_in[2];
  const float* lambda0 = (const float*)d_in[3];
  const float* pemb    = (const float*)d_in[4];   // [2,14,1]
  const int*   indices = (const int*)d_in[5];     // [2, E]
  const int*   pt      = (const int*)d_in[6];
  const int* row = indices;
  const int* col = indices + NEDGES;
  float* out = (float*)d_out;

  // workspace carve-out (256B aligned slots)
  char* p = (char*)d_ws;
  auto carve = [&](size_t bytes) -> char* {
    char* r = p;
    p += (bytes + 255) & ~(size_t)255;
    return r;
  };
  float*    y    = (float*)   carve((size_t)NNODES * HIDDEN * 4);
  float*    emb1 = (float*)   carve((size_t)NNODES * HIDDEN * 4);
  float*    s0a  = (float*)   carve((size_t)NEDGES * 4);  // s0, then reused as 'a'
  float*    e0   = (float*)   carve((size_t)NEDGES * 4);
  char*     statsBase = p;
  unsigned* m0u  = (unsigned*)carve((size_t)NNODES * 4);
  unsigned* m1u  = (unsigned*)carve((size_t)NNODES * 4);
  float*    d0   = (float*)   carve((size_t)NNODES * 4);
  float*    d1   = (float*)   carve((size_t)NNODES * 4);
  size_t statsBytes = (size_t)(p - statsBase);

  const dim3 B(256);
  const int GRID_NODE_ELEM = (NNODES * HIDDEN) / 256;        // 75000
  const int GRID_LN        = ((NNODES / 16) + 7) / 8;        // 2344 (8 waves * 16 nodes)
  const int GRID_SCORE     = NEDGES / 16;                    // 78125
  const int GRID_EDGE      = (NEDGES + 255) / 256;           // 4883
  const int GRID_AGG       = (NEDGES * HIDDEN) / 256;        // 312500

  // out = emb0
  hipLaunchKernelGGL(init_out_kernel, dim3(GRID_NODE_ELEM), B, 0, stream, ue, ie, out);

  for (int l = 0; l < 2; ++l) {
    hipMemsetAsync(statsBase, 0, statsBytes, stream);  // m0u,m1u (enc identity), d0,d1
    if (l == 0)
      hipMemsetAsync(emb1, 0, (size_t)NNODES * HIDDEN * 4, stream);

    if (l == 0)
      hipLaunchKernelGGL(ln_wmma_kernel, dim3(GRID_LN), B, 0, stream, ue, ie, NUSERS, y);
    else
      hipLaunchKernelGGL(ln_wmma_kernel, dim3(GRID_LN), B, 0, stream, emb1, emb1, NNODES, y);

    hipLaunchKernelGGL(score_kernel, dim3(GRID_SCORE), B, 0, stream,
                       y, eigs, row, col, pt, lambda0 + l, pemb + l * NPATHS,
                       s0a, m0u, m1u);
    hipLaunchKernelGGL(expsum_kernel, dim3(GRID_EDGE), B, 0, stream,
                       s0a, row, pt, pemb + l * NPATHS, m0u, m1u, e0, d0, d1);
    hipLaunchKernelGGL(coef_kernel, dim3(GRID_EDGE), B, 0, stream,
                       row, pt, pemb + l * NPATHS, m1u, e0, d0, d1, s0a);

    float* target = (l == 0) ? emb1 : out;  // last layer scatters straight into out
    hipLaunchKernelGGL(aggregate_kernel, dim3(GRID_AGG), B, 0, stream,
                       s0a, y, row, col, target);

    if (l == 0)
      hipLaunchKernelGGL(add_kernel, dim3(GRID_NODE_ELEM), B, 0, stream, out, emb1);
  }

  hipLaunchKernelGGL(scale_kernel, dim3(GRID_NODE_ELEM), B, 0, stream, out, 1.0f / 3.0f);
}